// DCTEncoderLayer_79061757985121
// MI455X (gfx1250) — compile-verified
//
#include <hip/hip_runtime.h>

typedef __attribute__((ext_vector_type(2))) float v2f;
typedef __attribute__((ext_vector_type(4))) float v4f;
typedef __attribute__((ext_vector_type(8))) float v8f;

#define IMGB   32
#define IMGH   512
#define IMGW   512
#define BS     32
#define NHW    16                      // blocks per image side
#define MDIM   1024                    // number of DCT coefficients
#define KDIM   1024                    // pixels per block
#define NCOL   (IMGB * 3 * NHW * NHW)  // 24576 block-columns

// ---------------------------------------------------------------------------
// Pass 1: RGB -> YCbCr (rescaled to [-1,1]) + blockize into column-major
// B-matrix  Bmat[col][pix],  col = (b*3+c)*256 + m*16+n,  pix = yin*32+xin.
// One thread per pixel; a wave32 covers one 32-wide block row -> contiguous
// 128B global writes per channel, contiguous 128B reads of each plane.
// ---------------------------------------------------------------------------
__global__ __launch_bounds__(256) void ycbcr_blockize_kernel(
    const float* __restrict__ rgb, float* __restrict__ Bmat) {
  unsigned tid = blockIdx.x * 256u + threadIdx.x;   // 0 .. 32*512*512-1
  int x = tid & 511;
  int y = (tid >> 9) & 511;
  int b = tid >> 18;

  const size_t plane = (size_t)IMGH * IMGW;
  size_t base = (size_t)b * 3 * plane + (size_t)y * IMGW + x;
  float r  = rgb[base];
  float g  = rgb[base + plane];
  float bl = rgb[base + 2 * plane];

  float yv = 0.299f * r + 0.587f * g + 0.114f * bl;
  float cb = (bl - yv) * 0.564f + 0.5f;
  float cr = (r  - yv) * 0.713f + 0.5f;
  yv = 2.0f * yv - 1.0f;
  cb = 2.0f * cb - 1.0f;
  cr = 2.0f * cr - 1.0f;

  int m = y >> 5, yin = y & 31;
  int n = x >> 5, xin = x & 31;
  int pix = yin * BS + xin;
  size_t col0 = ((size_t)b * 3) * 256 + (m * 16 + n);

  Bmat[(col0        ) * KDIM + pix] = yv;
  Bmat[(col0 + 256u ) * KDIM + pix] = cb;
  Bmat[(col0 + 512u ) * KDIM + pix] = cr;
}

// ---------------------------------------------------------------------------
// Pass 2: C[m][col] = sum_k kernels[m][k] * Bmat[col][k], scaled by
// 0.0625*factors[m], scattered into the (B, 3*1024, 16, 16) output.
// Full-precision f32 WMMA (v_wmma_f32_16x16x4_f32).
// 256 threads = 8 waves; tile BM=128 x BN=64 x BK=32; each wave: 32x32 C.
// ---------------------------------------------------------------------------
#define BM 128
#define BN 64
#define BK 32
#define LDA (BK + 2)  // 34: conflict-free float2 fragment loads
#define LDB (BN + 1)  // 65

__global__ __launch_bounds__(256) void dct_wmma_gemm_kernel(
    const float* __restrict__ A,        // kernels, [1024][1024] row-major
    const float* __restrict__ Bmat,     // [NCOL][1024] (col-major blocks)
    const float* __restrict__ factors,  // [1024]
    float* __restrict__ out) {
  __shared__ __align__(16) float As[BM * LDA];
  __shared__ __align__(16) float Bs[BK * LDB];

  const int tid   = threadIdx.x;
  const int Mbase = blockIdx.x * BM;   // grid.x = 8  (M varies fastest ->
  const int Nbase = blockIdx.y * BN;   //  concurrent WGs share B panel in L2)
  const int lane  = tid & 31;
  const int wave  = tid >> 5;
  const int wm    = wave & 3;          // wave M slot (0..3) -> 32 rows
  const int wn    = wave >> 2;         // wave N slot (0..1) -> 32 cols
  const int m16   = lane & 15;         // row/col within 16x16 fragment
  const int kh    = lane >> 4;         // K-half select (ISA A/B layout)

  v8f c[2][2] = {};

  for (int k0 = 0; k0 < KDIM; k0 += BK) {
    // --- stage A tile [BM][BK] as As[m][k]: float4 global, float2 LDS ---
    {
      int koff = (tid & 7) * 4;        // 8 threads per 32-wide row
      int row0 = tid >> 3;             // 0..31, step 32 -> 128 rows
#pragma unroll
      for (int r = 0; r < 4; ++r) {
        int row = row0 + r * 32;
        v4f a = *(const v4f*)(A + (size_t)(Mbase + row) * KDIM + k0 + koff);
        float* dst = &As[row * LDA + koff];
        *(v2f*)(dst)     = (v2f){a.x, a.y};
        *(v2f*)(dst + 2) = (v2f){a.z, a.w};
      }
    }
    // --- stage B tile [BK][BN] as Bs[k][n]: float4 along pix (contiguous) ---
    {
      int koff = (tid & 7) * 4;
      int col0 = tid >> 3;             // 0..31, step 32 -> 64 cols
#pragma unroll
      for (int r = 0; r < 2; ++r) {
        int col = col0 + r * 32;
        v4f b = *(const v4f*)(Bmat + (size_t)(Nbase + col) * KDIM + k0 + koff);
        Bs[(koff + 0) * LDB + col] = b.x;
        Bs[(koff + 1) * LDB + col] = b.y;
        Bs[(koff + 2) * LDB + col] = b.z;
        Bs[(koff + 3) * LDB + col] = b.w;
      }
    }
    __syncthreads();

#pragma unroll
    for (int kk = 0; kk < BK; kk += 4) {
      int ka = kk + 2 * kh;            // lanes 0-15: K={kk,kk+1}; 16-31: +2
      v2f aF[2], bF[2];
#pragma unroll
      for (int i = 0; i < 2; ++i)
        aF[i] = *(const v2f*)&As[(wm * 32 + i * 16 + m16) * LDA + ka];
#pragma unroll
      for (int j = 0; j < 2; ++j) {
        int n = wn * 32 + j * 16 + m16;
        bF[j] = (v2f){Bs[ka * LDB + n], Bs[(ka + 1) * LDB + n]};
      }
#pragma unroll
      for (int i = 0; i < 2; ++i)
#pragma unroll
        for (int j = 0; j < 2; ++j)
          c[i][j] = __builtin_amdgcn_wmma_f32_16x16x4_f32(
              false, aF[i], false, bF[j], (short)0, c[i][j], false, false);
    }
    __syncthreads();
  }

  // --- epilogue: scale by (2/32)*factors[m] and scatter ---
  // C/D layout: VGPR v, lanes 0-15 -> M=v, N=lane; lanes 16-31 -> M=v+8.
#pragma unroll
  for (int i = 0; i < 2; ++i) {
#pragma unroll
    for (int j = 0; j < 2; ++j) {
      int col = Nbase + wn * 32 + j * 16 + m16;
      size_t obase = (size_t)(col >> 8) * (MDIM * 256) + (col & 255);
#pragma unroll
      for (int v = 0; v < 8; ++v) {
        int kcoef = Mbase + wm * 32 + i * 16 + v + 8 * kh;
        float s = 0.0625f * factors[kcoef];
        out[obase + (size_t)kcoef * 256] = c[i][j][v] * s;
      }
    }
  }
}

// ---------------------------------------------------------------------------
extern "C" void kernel_launch(void* const* d_in, const int* in_sizes, int n_in,
                              void* d_out, int out_size, void* d_ws,
                              size_t ws_size, hipStream_t stream) {
  const float* rgb     = (const float*)d_in[0];  // (32,3,512,512) f32
  const float* kernels = (const float*)d_in[1];  // (1024,32,32)   f32
  const float* factors = (const float*)d_in[2];  // (1024,)        f32
  float* out  = (float*)d_out;                   // (32,3072,16,16) f32
  float* Bmat = (float*)d_ws;                    // [NCOL][1024] staging (~96 MB)

  // Pass 1: 32*512*512 pixels, 256 threads each
  ycbcr_blockize_kernel<<<(IMGB * IMGH * IMGW) / 256, 256, 0, stream>>>(rgb,
                                                                        Bmat);
  // Pass 2: WMMA GEMM, grid = (M tiles, N tiles) with M fastest
  dim3 grid(MDIM / BM, NCOL / BN);
  dct_wmma_gemm_kernel<<<grid, 256, 0, stream>>>(kernels, Bmat, factors, out);
}